// SpO2Net_936302870551
// MI455X (gfx1250) — compile-verified
//
#include <hip/hip_runtime.h>

// Ext-vector types (HIP's float4 is a struct; builtins need real vector types)
typedef float f4  __attribute__((ext_vector_type(4)));
typedef float v2f __attribute__((ext_vector_type(2)));
typedef float v8f __attribute__((ext_vector_type(8)));

static constexpr int ROW_LEN = 4096;   // L
static constexpr int TPB     = 256;    // 8 wave32 per block
static constexpr int F4S     = ROW_LEN / (TPB * 4);  // 4 float4 per thread

// out[row, l] = y[row, l] * (sum_x[row] + sum_y[row]) / L
// One block per row. Streaming: each global byte touched exactly once (NT hints).
__global__ __launch_bounds__(TPB) void spo2_rowmean_scale(
    const float* __restrict__ x,
    const float* __restrict__ y,
    float* __restrict__ out)
{
    const int row = blockIdx.x;
    const int tid = threadIdx.x;

    const f4* xr = (const f4*)(x   + (size_t)row * ROW_LEN);
    const f4* yr = (const f4*)(y   + (size_t)row * ROW_LEN);
    f4*       orw = (f4*)     (out + (size_t)row * ROW_LEN);

    // ---- Pass 1: coalesced b128 NT loads; keep y in registers; partial sum(x+y)
    f4 yv[F4S];
    float partial = 0.0f;
#pragma unroll
    for (int i = 0; i < F4S; ++i) {
        const int idx = i * TPB + tid;            // coalesced across the wave
        f4 xv = __builtin_nontemporal_load(xr + idx);
        yv[i] = __builtin_nontemporal_load(yr + idx);
        partial += (xv.x + xv.y) + (xv.z + xv.w);
        partial += (yv[i].x + yv[i].y) + (yv[i].z + yv[i].w);
    }

    __shared__ float lds_p[TPB];
    __shared__ float lds_total;
    lds_p[tid] = partial;
    __syncthreads();

    // ---- Block reduction of 256 partials on wave 0 via WMMA (A x ones):
    // D[m,n] = sum_k A[m,k]; chain 4 WMMAs over 4 chunks of 64 partials.
    // B = all-ones is layout-independent, so only the documented A and C/D
    // f32 layouts are relied upon.
    if (tid < 32) {                                // wave 0 fully active: EXEC all 1s
        const int lane  = tid;
        const int m     = lane & 15;               // A-matrix row
        const int khalf = lane >> 4;               // lanes 0-15: K=0,1; lanes 16-31: K=2,3
        v2f b;  b.x = 1.0f; b.y = 1.0f;            // all-ones B (4x16)
        v8f acc = {};                              // C = 0
#pragma unroll
        for (int j = 0; j < 4; ++j) {
            const int base = 64 * j + m * 4 + 2 * khalf;
            v2f a;
            a.x = lds_p[base + 0];
            a.y = lds_p[base + 1];
            // (neg_a, A, neg_b, B, c_mod, C, reuse_a, reuse_b)
            acc = __builtin_amdgcn_wmma_f32_16x16x4_f32(
                false, a, false, b, (short)0, acc, false, false);
        }
        // D vgpr r: lanes 0-15 -> M=r, lanes 16-31 -> M=r+8 (N replicated).
        float s = ((acc[0] + acc[1]) + (acc[2] + acc[3]))
                + ((acc[4] + acc[5]) + (acc[6] + acc[7]));
        s += __shfl_xor(s, 16, 32);                // combine M=0..7 with M=8..15
        if (lane == 0) lds_total = s;
    }
    __syncthreads();

    // ---- Pass 2: scale registered y, NT b128 stores (pure streaming write)
    const float scale = lds_total * (1.0f / (float)ROW_LEN);
#pragma unroll
    for (int i = 0; i < F4S; ++i) {
        const int idx = i * TPB + tid;
        f4 o = yv[i] * scale;
        __builtin_nontemporal_store(o, orw + idx);
    }
}

extern "C" void kernel_launch(void* const* d_in, const int* in_sizes, int n_in,
                              void* d_out, int out_size, void* d_ws, size_t ws_size,
                              hipStream_t stream) {
    (void)n_in; (void)out_size; (void)d_ws; (void)ws_size;
    const float* x = (const float*)d_in[0];
    const float* y = (const float*)d_in[1];
    float* out = (float*)d_out;

    const int rows = in_sizes[0] / ROW_LEN;        // B*C = 2304
    spo2_rowmean_scale<<<rows, TPB, 0, stream>>>(x, y, out);
}